// GNN_33440615367233
// MI455X (gfx1250) — compile-verified
//
#include <hip/hip_runtime.h>

// GCN (2× GCNConv + linear + sigmoid) for MI455X (gfx1250, wave32).
// Memory-bound scatter/gather workload; GEMMs done with V_WMMA_F32_16X16X4_F32
// (fp32-exact), scatter with native global_atomic_add_f32.

typedef __attribute__((ext_vector_type(2))) float v2f;
typedef __attribute__((ext_vector_type(8))) float v8f;

#define WPB 8  // waves per block (256 threads)

__device__ __forceinline__ void atomAdd(float* p, float v) {
  unsafeAtomicAdd(p, v);  // -> global_atomic_add_f32 (no CAS loop)
}

// ---------------- degree / normalization ----------------
__global__ void k_deg_init(unsigned* __restrict__ deg, int n) {
  int i = blockIdx.x * blockDim.x + threadIdx.x;
  if (i < n) deg[i] = 1u;  // self-loop contribution
}

__global__ void k_deg_edges(const long long* __restrict__ ei, unsigned* __restrict__ deg, int E) {
  int e = blockIdx.x * blockDim.x + threadIdx.x;
  if (e < E) atomicAdd(&deg[(int)ei[(size_t)E + e]], 1u);
}

__global__ void k_dis(float* __restrict__ dis, int n) {  // in place: uint count -> rsqrt
  int i = blockIdx.x * blockDim.x + threadIdx.x;
  if (i < n) {
    float d = (float)((const unsigned*)dis)[i];  // >= 1 always (self-loop)
    dis[i] = rsqrtf(d);
  }
}

// ---------------- layer 1 GEMM: y1 = (x @ W1) * dis, acc1 = y1 ----------------
// One wave computes a 16x16 tile via a single V_WMMA_F32_16X16X4_F32.
__global__ void k_gemm1(const float* __restrict__ x, const float* __restrict__ W1,
                        const float* __restrict__ dis, float* __restrict__ y1,
                        float* __restrict__ acc1, int ntiles) {
  const int lane = threadIdx.x & 31;
  const int tile = blockIdx.x * WPB + (threadIdx.x >> 5);
  if (tile >= ntiles) return;  // wave-uniform: EXEC all-ones inside
  const int row0 = tile << 4;
  const int m = lane & 15;                 // M (and N for B/C fragments)
  const int kh = (lane >= 16) ? 2 : 0;     // K half select
  // A fragment: A[m][kh], A[m][kh+1]  (x row-major, 4 floats per row)
  const float* xr = x + (size_t)(row0 + m) * 4 + kh;
  v2f a; a.x = xr[0]; a.y = xr[1];
  // B fragment: B[kh][n], B[kh+1][n]  (W1 is 4x16 row-major)
  v2f b; b.x = W1[kh * 16 + m]; b.y = W1[(kh + 1) * 16 + m];
  v8f c = {};
  c = __builtin_amdgcn_wmma_f32_16x16x4_f32(false, a, false, b, (short)0, c, false, false);
#pragma unroll
  for (int r = 0; r < 8; ++r) {  // D: VGPR r holds M=r (lanes 0-15) / M=r+8 (lanes 16-31)
    int node = row0 + r + ((lane >= 16) ? 8 : 0);
    float v = c[r] * dis[node];
    y1[(size_t)node * 16 + m]  = v;
    acc1[(size_t)node * 16 + m] = v;  // self-loop seed
  }
}

// ---------------- layer 1 scatter: acc1[dst] += y1[src] ----------------
__global__ void k_scatter1(const long long* __restrict__ ei, const float* __restrict__ y1,
                           float* __restrict__ acc1, int E) {
  int e = blockIdx.x * blockDim.x + threadIdx.x;
  if (e >= E) return;
  int s = (int)ei[e];
  int d = (int)ei[(size_t)E + e];
  const float4* ys = (const float4*)(y1 + (size_t)s * 16);
  float4 v0 = ys[0], v1 = ys[1], v2 = ys[2], v3 = ys[3];
  float* ad = acc1 + (size_t)d * 16;
  atomAdd(ad + 0,  v0.x); atomAdd(ad + 1,  v0.y); atomAdd(ad + 2,  v0.z); atomAdd(ad + 3,  v0.w);
  atomAdd(ad + 4,  v1.x); atomAdd(ad + 5,  v1.y); atomAdd(ad + 6,  v1.z); atomAdd(ad + 7,  v1.w);
  atomAdd(ad + 8,  v2.x); atomAdd(ad + 9,  v2.y); atomAdd(ad + 10, v2.z); atomAdd(ad + 11, v2.w);
  atomAdd(ad + 12, v3.x); atomAdd(ad + 13, v3.y); atomAdd(ad + 14, v3.z); atomAdd(ad + 15, v3.w);
}

// ---------------- layer 2 GEMM: h1 = relu(dis*acc1 + b1); y2 = (h1 @ W2)*dis ----------------
// Per wave: 16x32 output tile = 2 N-halves x 4 chained K-steps of 16x16x4 WMMA.
__global__ void k_gemm2(const float* __restrict__ acc1, const float* __restrict__ bias1,
                        const float* __restrict__ W2, const float* __restrict__ dis,
                        float* __restrict__ y2, float* __restrict__ acc2, int ntiles) {
  const int lane = threadIdx.x & 31;
  const int tile = blockIdx.x * WPB + (threadIdx.x >> 5);
  if (tile >= ntiles) return;
  const int row0 = tile << 4;
  const int m = lane & 15;
  const int kh = (lane >= 16) ? 2 : 0;
  const float dm = dis[row0 + m];
  v8f c0 = {}; v8f c1 = {};
#pragma unroll
  for (int s = 0; s < 4; ++s) {
    const int k0 = 4 * s + kh;
    // A fragment with bias+ReLU fused (recomputed per lane; trivially cheap)
    const float* ar = acc1 + (size_t)(row0 + m) * 16 + k0;
    v2f a;
    a.x = fmaxf(fmaf(dm, ar[0], bias1[k0]),     0.0f);
    a.y = fmaxf(fmaf(dm, ar[1], bias1[k0 + 1]), 0.0f);
    // B fragments for columns [0,16) and [16,32)  (W2 is 16x32 row-major)
    v2f bA; bA.x = W2[k0 * 32 + m];      bA.y = W2[(k0 + 1) * 32 + m];
    v2f bB; bB.x = W2[k0 * 32 + 16 + m]; bB.y = W2[(k0 + 1) * 32 + 16 + m];
    c0 = __builtin_amdgcn_wmma_f32_16x16x4_f32(false, a, false, bA, (short)0, c0, false, false);
    c1 = __builtin_amdgcn_wmma_f32_16x16x4_f32(false, a, false, bB, (short)0, c1, false, false);
  }
#pragma unroll
  for (int r = 0; r < 8; ++r) {
    int node = row0 + r + ((lane >= 16) ? 8 : 0);
    float sc = dis[node];
    float va = c0[r] * sc, vb = c1[r] * sc;
    size_t base = (size_t)node * 32;
    y2[base + m]       = va;  acc2[base + m]       = va;
    y2[base + 16 + m]  = vb;  acc2[base + 16 + m]  = vb;
  }
}

// ---------------- layer 2 scatter ----------------
__global__ void k_scatter2(const long long* __restrict__ ei, const float* __restrict__ y2,
                           float* __restrict__ acc2, int E) {
  int e = blockIdx.x * blockDim.x + threadIdx.x;
  if (e >= E) return;
  int s = (int)ei[e];
  int d = (int)ei[(size_t)E + e];
  const float4* ys = (const float4*)(y2 + (size_t)s * 32);
  float* ad = acc2 + (size_t)d * 32;
#pragma unroll
  for (int q = 0; q < 8; ++q) {
    float4 v = ys[q];
    atomAdd(ad + 4 * q + 0, v.x);
    atomAdd(ad + 4 * q + 1, v.y);
    atomAdd(ad + 4 * q + 2, v.z);
    atomAdd(ad + 4 * q + 3, v.w);
  }
}

// ---------------- final: sigmoid(relu(dis*acc2 + b2) @ Wl + bl) ----------------
__global__ void k_final(const float* __restrict__ acc2, const float* __restrict__ bias2,
                        const float* __restrict__ dis, const float* __restrict__ Wl,
                        const float* __restrict__ bl, float* __restrict__ out, int n) {
  int i = blockIdx.x * blockDim.x + threadIdx.x;
  if (i >= n) return;
  float s = dis[i];
  const float* row = acc2 + (size_t)i * 32;
  float z = bl[0];
#pragma unroll
  for (int c = 0; c < 32; ++c) {
    float h = fmaxf(fmaf(s, row[c], bias2[c]), 0.0f);
    z = fmaf(h, Wl[c], z);
  }
  out[i] = 1.0f / (1.0f + expf(-z));
}

extern "C" void kernel_launch(void* const* d_in, const int* in_sizes, int n_in,
                              void* d_out, int out_size, void* d_ws, size_t ws_size,
                              hipStream_t stream) {
  const float*     x   = (const float*)d_in[0];
  const long long* ei  = (const long long*)d_in[1];  // int64 [2, E]
  const float*     W1  = (const float*)d_in[2];
  const float*     b1  = (const float*)d_in[3];
  const float*     W2  = (const float*)d_in[4];
  const float*     b2  = (const float*)d_in[5];
  const float*     Wl  = (const float*)d_in[6];
  const float*     bl  = (const float*)d_in[7];
  float* out = (float*)d_out;

  const int N = in_sizes[0] / 4;   // 100000 (multiple of 16)
  const int E = in_sizes[1] / 2;   // 2500000
  const int ntiles = N / 16;

  // Workspace carve-out (floats): dis | y1 | acc1 | y2 | acc2  (~38.8 MB total)
  float* ws   = (float*)d_ws;
  float* dis  = ws;                          // N (uint counts, then rsqrt floats)
  float* y1   = ws + (size_t)102400;         // N*16
  float* acc1 = y1 + (size_t)N * 16;         // N*16
  float* y2   = acc1 + (size_t)N * 16;       // N*32
  float* acc2 = y2 + (size_t)N * 32;         // N*32

  const int TB = 256;
  dim3 bn((N + TB - 1) / TB), be((E + TB - 1) / TB), bt((ntiles + WPB - 1) / WPB);

  k_deg_init <<<bn, TB, 0, stream>>>((unsigned*)dis, N);
  k_deg_edges<<<be, TB, 0, stream>>>(ei, (unsigned*)dis, E);
  k_dis      <<<bn, TB, 0, stream>>>(dis, N);
  k_gemm1    <<<bt, TB, 0, stream>>>(x, W1, dis, y1, acc1, ntiles);
  k_scatter1 <<<be, TB, 0, stream>>>(ei, y1, acc1, E);
  k_gemm2    <<<bt, TB, 0, stream>>>(acc1, b1, W2, dis, y2, acc2, ntiles);
  k_scatter2 <<<be, TB, 0, stream>>>(ei, y2, acc2, E);
  k_final    <<<bn, TB, 0, stream>>>(acc2, b2, dis, Wl, bl, out, N);
}